// CWTSimilarity_13417477832916
// MI455X (gfx1250) — compile-verified
//
#include <hip/hip_runtime.h>
#include <math.h>

typedef __attribute__((ext_vector_type(2))) float        v2f;
typedef __attribute__((ext_vector_type(8))) float        v8f;
typedef __attribute__((ext_vector_type(4))) unsigned int v4u;
typedef __attribute__((ext_vector_type(4))) int          v4i;
typedef __attribute__((ext_vector_type(8))) int          v8i;

#define WDIM   2048      // signal length
#define KTAPS  512       // padded filter taps (v = k - 256)
#define NSC    32        // padded scale count
#define SREAL  31        // real scales
#define GT_PAD 516       // LDS row stride in floats (512 + TDM pad 2dw per 256dw)
#define NF     95        // 3*31 + 2 features
#define FEPS   1e-8f

// ---------------------------------------------------------------------------
// Kernel 0: build filter bank G[32][512] on device (one-time, tiny).
// G_s[v] = -sqrt(s) * (filt_s[8s-v] - filt_s[8s-v-1]),  filt_s[i] = int_psi[j(i)]
// ---------------------------------------------------------------------------
__global__ __launch_bounds__(256) void cwt_filters_kernel(float* __restrict__ Gt) {
    __shared__ float int_psi[1024];
    const int tid = threadIdx.x;
    for (int i = tid; i < 1024; i += 256) {
        double xg = -8.0 + 16.0 * (double)i / 1023.0;
        int_psi[i] = (float)(exp(-xg * xg * 0.5) * cos(5.0 * xg));
    }
    __syncthreads();
    if (tid == 0) {  // cumsum * step (serial; 1024 adds, one-time)
        const double step = 16.0 / 1023.0;
        double acc = 0.0;
        for (int i = 0; i < 1024; ++i) {
            acc += (double)int_psi[i];
            int_psi[i] = (float)(acc * step);
        }
    }
    __syncthreads();
    const double step = 16.0 / 1023.0;
    for (int idx = tid; idx < NSC * KTAPS; idx += 256) {
        int sc = idx >> 9;
        int k  = idx & (KTAPS - 1);
        float g = 0.f;
        if (sc < SREAL) {
            int s = sc + 1;
            int L = 16 * s + 1;
            int v = k - 256;
            int t1 = 8 * s - v;
            int t0 = t1 - 1;
            float F1 = 0.f, F0 = 0.f;
            if (t1 >= 0 && t1 < L) {
                int j = (int)((double)(16 * s - t1) / ((double)s * step));
                F1 = int_psi[j];
            }
            if (t0 >= 0 && t0 < L) {
                int j = (int)((double)(16 * s - t0) / ((double)s * step));
                F0 = int_psi[j];
            }
            g = -sqrtf((float)s) * (F1 - F0);
        }
        Gt[idx] = g;
    }
}

// ---------------------------------------------------------------------------
// Kernel 1: fused CWT (WMMA f32 16x16x4, Toeplitz A from LDS) + features.
// Filter bank staged LDS-side by the Tensor Data Mover with LDS padding
// (2 dwords per 256 dwords -> 516-float row stride, bank-conflict-free).
// One workgroup (8 waves) per signal row; each wave computes BOTH scale
// tiles of a w-tile (2 accumulators, shared A fragments).
// ---------------------------------------------------------------------------
__global__ __launch_bounds__(256) void cwt_main_kernel(
    const float* __restrict__ x, const float* __restrict__ Gt,
    float* __restrict__ cwt_out, float* __restrict__ feat_out) {

    __shared__ float xpad[WDIM + 512];          // x row, zero-padded 256 each side
    __shared__ float GtL[NSC * GT_PAD];         // filter bank (TDM-padded rows)
    __shared__ float sum_p[WDIM];               // sum over scales of p at each w
    __shared__ float sum_sp[WDIM];              // sum over scales of s*p at each w
    __shared__ float scaleA[NSC];               // sum |c| per scale
    __shared__ float scaleA2[NSC];              // sum c^2 per scale
    __shared__ float red[256];
    __shared__ float red2[256];

    const int m    = blockIdx.x;
    const int tid  = threadIdx.x;
    const int lane = tid & 31;
    const int wv   = tid >> 5;

    // ---- TDM: DMA the 64KB filter bank global->LDS with padding ----
    if (wv == 0) {
        unsigned lds_base = (unsigned)(unsigned long long)(void*)&GtL[0];
        unsigned long long ga = (unsigned long long)Gt;
        v4u g0;
        g0.x = 0x1u;                                   // count=1 (valid descriptor)
        g0.y = lds_base;                               // lds_addr (bytes)
        g0.z = (unsigned)(ga & 0xFFFFFFFFull);         // global_addr[31:0]
        g0.w = (unsigned)((ga >> 32) & 0x01FFFFFFull)  // global_addr[56:32]
             | 0x80000000u;                            // type=2 ("image")
        v8i g1;
        g1[0] = (2 << 16)      // data_size = 4B
              | (1 << 20)      // pad_enable
              | (7 << 22)      // pad_interval code 7 = every 256 dwords
              | (1 << 25);     // pad_amount  code 1 = 2 dwords
        g1[1] = (int)((16384u & 0xFFFFu) << 16);       // tensor_dim0[15:0]
        g1[2] = (int)((16384u >> 16) | (1u << 16));    // tensor_dim0[31:16] | tensor_dim1 lo
        g1[3] = (int)((16384u & 0xFFFFu) << 16);       // tensor_dim1 hi | tile_dim0
        g1[4] = 1;                                     // tile_dim1 = 1, tile_dim2 = 0
        g1[5] = 16384;                                 // tensor_dim0_stride lo
        g1[6] = 0;
        g1[7] = 0;
        v4i zz = {0, 0, 0, 0};
#if defined(__clang_major__) && (__clang_major__ >= 23)
        v8i zz8 = {0, 0, 0, 0, 0, 0, 0, 0};
        __builtin_amdgcn_tensor_load_to_lds(g0, g1, zz, zz, zz8, 0);
#else
        __builtin_amdgcn_tensor_load_to_lds(g0, g1, zz, zz, 0);
#endif
    }

    // zero pad regions (disjoint from payload fill; no barrier needed)
    xpad[tid] = 0.f;
    xpad[WDIM + 256 + tid] = 0.f;
    if (tid < NSC) { scaleA[tid] = 0.f; scaleA2[tid] = 0.f; }

    // stage x row into LDS (vectorized global loads)
    {
        const float4* xr = (const float4*)(x + (size_t)m * WDIM);
        for (int i = tid; i < WDIM / 4; i += 256)
            *((float4*)&xpad[256 + i * 4]) = xr[i];
    }
    if (wv == 0) __builtin_amdgcn_s_wait_tensorcnt(0);  // TDM writes visible
    __syncthreads();

    // WMMA fragment roles (wave32):
    //  A (16x4 f32): M = lane&15, K = (lane>>4)*2 + {0,1}
    //  B (4x16 f32): N = lane&15, K = (lane>>4)*2 + {0,1}
    //  C/D: lanes 0-15 N=lane, vgpr r -> M=r; lanes 16-31 -> M=r+8
    const int row   = lane & 15;
    const int khalf = (lane >> 4) << 1;
    const int sc0   = row;            // scales 1..16
    const int sc1   = row + 16;       // scales 17..32 (32 = zero pad)
    float sA0 = 0.f, sA20 = 0.f, sA1 = 0.f, sA21 = 0.f;

    for (int tile = wv; tile < 128; tile += 8) {   // 128 w-tiles; both s-tiles
        const int w0 = tile << 4;
        const float* ax  = &xpad[w0 + row + khalf];
        const float* b0p = &GtL[sc0 * GT_PAD + khalf];
        const float* b1p = &GtL[sc1 * GT_PAD + khalf];

        v8f acc0 = {}, acc1 = {};
        // k in [0,256): LDS index = sc*516 + k
        for (int k0 = 0; k0 < 256; k0 += 4) {
            v2f a, b0, b1;
            a.x  = ax[k0];      a.y  = ax[k0 + 1];
            b0.x = b0p[k0];     b0.y = b0p[k0 + 1];
            b1.x = b1p[k0];     b1.y = b1p[k0 + 1];
            acc0 = __builtin_amdgcn_wmma_f32_16x16x4_f32(
                false, a, false, b0, (short)0, acc0, false, false);
            acc1 = __builtin_amdgcn_wmma_f32_16x16x4_f32(
                false, a, false, b1, (short)0, acc1, false, false);
        }
        // k in [256,512): TDM pad shifts row interior by +2 dwords
        for (int k0 = 256; k0 < 512; k0 += 4) {
            v2f a, b0, b1;
            a.x  = ax[k0];          a.y  = ax[k0 + 1];
            b0.x = b0p[k0 + 2];     b0.y = b0p[k0 + 3];
            b1.x = b1p[k0 + 2];     b1.y = b1p[k0 + 3];
            acc0 = __builtin_amdgcn_wmma_f32_16x16x4_f32(
                false, a, false, b0, (short)0, acc0, false, false);
            acc1 = __builtin_amdgcn_wmma_f32_16x16x4_f32(
                false, a, false, b1, (short)0, acc1, false, false);
        }

        const int wbase = w0 + ((lane >> 4) << 3);  // first w this lane covers
        #pragma unroll
        for (int r = 0; r < 8; ++r) {
            float c0 = acc0[r], c1 = acc1[r];
            float p0 = c0 * c0, p1 = c1 * c1;
            sA0 += fabsf(c0); sA20 += p0;
            sA1 += fabsf(c1); sA21 += p1;
            // this lane covers scales sc0 and sc1 at the same w
            float p  = p0 + p1;
            float sp = p0 * (float)(sc0 + 1) + p1 * (float)(sc1 + 1);
            #pragma unroll
            for (int mk = 1; mk < 16; mk <<= 1) {   // all 32 scales per half-wave
                p  += __shfl_xor(p,  mk, 16);
                sp += __shfl_xor(sp, mk, 16);
            }
            if ((lane & 15) == 0) {                 // unique writer per w
                sum_p[wbase + r]  = p;
                sum_sp[wbase + r] = sp;
            }
        }

        // write cwt tiles: 8 contiguous w per lane -> b128 stores
        {
            float* o0 = cwt_out + ((size_t)m * SREAL + sc0) * WDIM + wbase;
            ((float4*)o0)[0] = make_float4(acc0[0], acc0[1], acc0[2], acc0[3]);
            ((float4*)o0)[1] = make_float4(acc0[4], acc0[5], acc0[6], acc0[7]);
            if (sc1 < SREAL) {
                float* o1 = cwt_out + ((size_t)m * SREAL + sc1) * WDIM + wbase;
                ((float4*)o1)[0] = make_float4(acc1[0], acc1[1], acc1[2], acc1[3]);
                ((float4*)o1)[1] = make_float4(acc1[4], acc1[5], acc1[6], acc1[7]);
            }
        }
    }

    atomicAdd(&scaleA [sc0], sA0);
    atomicAdd(&scaleA2[sc0], sA20);
    atomicAdd(&scaleA [sc1], sA1);
    atomicAdd(&scaleA2[sc1], sA21);
    __syncthreads();

    float* fo = feat_out + (size_t)m * NF;
    if (tid < SREAL) {
        float sa  = scaleA[tid];
        float sa2 = scaleA2[tid];
        float mean = sa  * (1.f / (float)WDIM);
        float ex2  = sa2 * (1.f / (float)WDIM);
        fo[tid]             = mean;              // mean_f
        fo[SREAL + tid]     = ex2 - mean * mean; // var_f (ddof=0)
        fo[2 * SREAL + tid] = sa2;               // energy
    }

    // centroid mean/std over w
    float lc = 0.f, lc2 = 0.f;
    for (int i = tid; i < WDIM; i += 256) {
        float c = sum_sp[i] / (sum_p[i] + FEPS);
        lc += c; lc2 += c * c;
    }
    red[tid] = lc; red2[tid] = lc2;
    __syncthreads();
    for (int off = 128; off > 0; off >>= 1) {
        if (tid < off) { red[tid] += red[tid + off]; red2[tid] += red2[tid + off]; }
        __syncthreads();
    }
    if (tid == 0) {
        float cm   = red[0]  * (1.f / (float)WDIM);
        float cvar = red2[0] * (1.f / (float)WDIM) - cm * cm;
        fo[93] = cm;
        fo[94] = sqrtf(fmaxf(cvar, 0.f));
    }
}

// ---------------------------------------------------------------------------
// Kernel 2a: L2-normalize feature rows, write padded [2048][96] to workspace.
// ---------------------------------------------------------------------------
__global__ __launch_bounds__(256) void feat_norm_kernel(
    const float* __restrict__ feat, float* __restrict__ fn) {
    int r    = blockIdx.x * 8 + (threadIdx.x >> 5);
    int lane = threadIdx.x & 31;
    const float* f = feat + (size_t)r * NF;
    float vals[3];
    float s = 0.f;
    #pragma unroll
    for (int i = 0; i < 3; ++i) {
        int idx = lane + 32 * i;
        float v = (idx < NF) ? f[idx] : 0.f;
        vals[i] = v;
        s += v * v;
    }
    #pragma unroll
    for (int mk = 16; mk; mk >>= 1) s += __shfl_xor(s, mk, 32);
    float nrm = sqrtf(s);
    float inv = (nrm == 0.f) ? 1.f : 1.f / nrm;   // sklearn semantics
    float* o = fn + (size_t)r * 96;
    #pragma unroll
    for (int i = 0; i < 3; ++i) {
        int idx = lane + 32 * i;
        if (idx < 96) o[idx] = (idx < NF) ? vals[i] * inv : 0.f;
    }
}

// ---------------------------------------------------------------------------
// Kernel 2b: cosine similarity GEMM per batch, WMMA f32 16x16x4, K=96.
// One wave per 16x16 output tile; 8 batches x 256 tiles = 2048 tiles.
// ---------------------------------------------------------------------------
__global__ __launch_bounds__(256) void sim_kernel(
    const float* __restrict__ fn, float* __restrict__ sim) {
    int t     = blockIdx.x * 8 + (threadIdx.x >> 5);
    int lane  = threadIdx.x & 31;
    int batch = t >> 8;
    int ti    = (t >> 4) & 15;
    int tj    = t & 15;
    int khalf = (lane >> 4) << 1;

    const float* A = fn + (size_t)(batch * 256 + ti * 16 + (lane & 15)) * 96 + khalf;
    const float* B = fn + (size_t)(batch * 256 + tj * 16 + (lane & 15)) * 96 + khalf;

    v8f acc = {};
    for (int k0 = 0; k0 < 96; k0 += 4) {
        v2f a, b;
        a.x = A[k0]; a.y = A[k0 + 1];
        b.x = B[k0]; b.y = B[k0 + 1];
        acc = __builtin_amdgcn_wmma_f32_16x16x4_f32(
            false, a, false, b, (short)0, acc, false, false);
    }
    int mrow = ti * 16 + ((lane >> 4) << 3);
    float* o = sim + (size_t)batch * 256 * 256 + (size_t)mrow * 256 + tj * 16 + (lane & 15);
    #pragma unroll
    for (int r = 0; r < 8; ++r) o[(size_t)r * 256] = acc[r];
}

// ---------------------------------------------------------------------------
extern "C" void kernel_launch(void* const* d_in, const int* in_sizes, int n_in,
                              void* d_out, int out_size, void* d_ws, size_t ws_size,
                              hipStream_t stream) {
    (void)in_sizes; (void)n_in; (void)out_size; (void)ws_size;
    const float* x = (const float*)d_in[0];

    // outputs concatenated flat: sim [8,256,256], feats [8,256,95], cwt [8,256,31,2048]
    float* sim  = (float*)d_out;
    float* feat = sim + (size_t)8 * 256 * 256;
    float* cwt  = feat + (size_t)8 * 256 * NF;

    float* Gt = (float*)d_ws;              // [32*512] filter bank (contiguous)
    float* fn = Gt + NSC * KTAPS;          // [2048*96] normalized feats

    cwt_filters_kernel<<<1, 256, 0, stream>>>(Gt);
    cwt_main_kernel<<<2048, 256, 0, stream>>>(x, Gt, cwt, feat);
    feat_norm_kernel<<<256, 256, 0, stream>>>(feat, fn);
    sim_kernel<<<256, 256, 0, stream>>>(fn, sim);
}